// WealthSignalVPNModel_82995948028462
// MI455X (gfx1250) — compile-verified
//
#include <hip/hip_runtime.h>
#include <hip/hip_bf16.h>
#include <math.h>

// ---------------------------------------------------------------------------
// WealthSignalVPN on MI455X (gfx1250, wave32).
//
// overall[t] = mean_i( sin(w*time[t] + phase_i) + 0.8*noise[i,t] )
//              + 0.5 + 0.3*sin(wb*time[t])
// out[0:T)  = sin(0.5*overall[t]),  out[T:2T) = overall[t]
//
// Memory-bound: noise stream = 57.6 MB @ 23.3 TB/s  => ~2.5 us floor.
//  - angle addition removes the 14.4M per-element sins (-> 11.4K setup sins)
//  - node reduction on the fp32 matrix core V_WMMA_F32_16X16X4_F32
//    (A = ones 16x4), full fp32 accumulation
//  - all noise loads issued off one base address with immediate IOFFSETs
// ---------------------------------------------------------------------------

typedef __attribute__((ext_vector_type(2))) float v2f;
typedef __attribute__((ext_vector_type(8))) float v8f;

#define NUM_NODES   10000
#define T_STEPS     1440
#define N_CHUNKS    50                              // 50 chunks * 200 nodes
#define NODES_PER_CHUNK (NUM_NODES / N_CHUNKS)      // 200
#define KSTEPS      (NODES_PER_CHUNK / 4)           // 50 WMMA k-steps of 4 nodes
#define T_TILES     (T_STEPS / 16)                  // 90 tiles of 16 timesteps

// workspace layout (floats)
#define WS_NODE_SC   0                               // float2[NUM_NODES]
#define WS_T_SC      (2 * NUM_NODES)                 // float2[T_STEPS]
#define WS_PARTIAL   (2 * NUM_NODES + 2 * T_STEPS)   // float[N_CHUNKS*T_STEPS]

#define TWO_PI_F     6.28318530717958647692f
#define W_WEALTH     (TWO_PI_F / 24.0f)              // 2*pi*WAVE_FREQUENCY
#define PHASE_STEP   (TWO_PI_F / (float)NUM_NODES)
#define W_BRAIN      (TWO_PI_F * (10.0f / 3600.0f))  // 2*pi*BRAINWAVE_FREQUENCY
#define OPP_SCALE    0.8f
#define IR_AMP       0.5f
#define BRAIN_AMP    0.3f
#define ENC_KEY      0.5f

// --------------------------------------------------------------------------
// Kernel 0: one-time trig tables. nodeSC[n] = {sin(phi_n), cos(phi_n)},
// tSC[t] = {sin(w*time[t]), cos(w*time[t])}.  11,440 precise sin/cos total.
// --------------------------------------------------------------------------
__global__ __launch_bounds__(256)
void wealth_build_tables(const float* __restrict__ time_tensor,
                         float* __restrict__ ws) {
    const int i = blockIdx.x * blockDim.x + threadIdx.x;
    float* nodeSC = ws + WS_NODE_SC;
    float* tSC    = ws + WS_T_SC;
    if (i < NUM_NODES) {
        const float ph = (float)i * PHASE_STEP;
        nodeSC[2 * i]     = sinf(ph);
        nodeSC[2 * i + 1] = cosf(ph);
    }
    if (i < T_STEPS) {
        const float w = W_WEALTH * time_tensor[i];
        tSC[2 * i]     = sinf(w);
        tSC[2 * i + 1] = cosf(w);
    }
}

// --------------------------------------------------------------------------
// Kernel 1: one wave32 per block. blockIdx.x = t-tile (16 timesteps),
// blockIdx.y = node chunk (200 nodes). Per element:
//   val = sin(wt)*cos(phi) + cos(wt)*sin(phi) + 0.8*noise   (mul + 2 fma)
// Node reduction via fp32 WMMA with A = ones; D row 0 -> partial sums.
// --------------------------------------------------------------------------
__global__ __launch_bounds__(32)
void wealth_partial_wmma(const float* __restrict__ noise,
                         const float* __restrict__ ws_ro,
                         float* __restrict__ partial) {
    const int lane = threadIdx.x;        // 0..31
    const int col  = lane & 15;          // timestep column within tile
    const int half = lane >> 4;          // 0: k-slots {0,1}, 1: k-slots {2,3}
    const int t    = blockIdx.x * 16 + col;

    const float2 wsc = *(const float2*)(ws_ro + WS_T_SC + 2 * t); // {sin,cos}(w*t)
    const float st = wsc.x, ct = wsc.y;

    v2f a; a.x = 1.0f; a.y = 1.0f;       // A = 16x4 all-ones
    v8f c = {};                          // C/D accumulator

    const int nbase = blockIdx.y * NODES_PER_CHUNK;

    // One base address per wave; all subsequent loads use compile-time
    // constant offsets (fit in the signed 24-bit IOFFSET).
    const float* __restrict__ p = noise
        + (size_t)(nbase + half * 2) * T_STEPS + t;          // this half's row 0
    const float* __restrict__ sctab = ws_ro + WS_NODE_SC
        + 2 * (nbase + half * 2);                            // 16B-aligned

    #pragma unroll
    for (int j = 0; j < KSTEPS; ++j) {
        // {sin,cos}(phi_n0), {sin,cos}(phi_n0+1) - uniform across each
        // 16-lane half (L0-resident); constant offset 32*j bytes.
        const float4 sc = *(const float4*)(sctab + 8 * j);
        const float nz0 = p[(size_t)j * 4 * T_STEPS];             // offset j*23040
        const float nz1 = p[(size_t)j * 4 * T_STEPS + T_STEPS];   // +5760

        v2f b;
        b.x = fmaf(st, sc.y, ct * sc.x) + OPP_SCALE * nz0;
        b.y = fmaf(st, sc.w, ct * sc.z) + OPP_SCALE * nz1;

        // D = ones(16x4) x B(4x16) + C : accumulate 4-node column sums in fp32
        c = __builtin_amdgcn_wmma_f32_16x16x4_f32(
                false, a, false, b, (short)0, c, false, false);
    }

    // D row 0 lives in VGPR0 of lanes 0..15 (all rows identical since A=ones).
    if (half == 0) {
        partial[blockIdx.y * T_STEPS + t] = c[0];
    }
}

// --------------------------------------------------------------------------
// Kernel 2: deterministic chunk reduction + pointwise finalize.
// --------------------------------------------------------------------------
__global__ __launch_bounds__(256)
void wealth_finalize(const float* __restrict__ time_tensor,
                     const float* __restrict__ partial,
                     float* __restrict__ out) {
    const int t = blockIdx.x * blockDim.x + threadIdx.x;
    if (t >= T_STEPS) return;

    float s = 0.0f;
    #pragma unroll 5
    for (int ch = 0; ch < N_CHUNKS; ++ch) {
        s += partial[ch * T_STEPS + t];
    }

    const float tt = time_tensor[t];
    const float brain = BRAIN_AMP * sinf(W_BRAIN * tt);
    const float overall = s * (1.0f / (float)NUM_NODES) + IR_AMP + brain;

    out[t]           = sinf(ENC_KEY * overall);   // encrypted_signal
    out[T_STEPS + t] = overall;                   // overall_signal
}

extern "C" void kernel_launch(void* const* d_in, const int* in_sizes, int n_in,
                              void* d_out, int out_size, void* d_ws, size_t ws_size,
                              hipStream_t stream) {
    const float* time_tensor = (const float*)d_in[0];   // [1440]
    const float* noise       = (const float*)d_in[1];   // [10000, 1440]
    float* out = (float*)d_out;                         // [2*1440]
    float* ws  = (float*)d_ws;                          // ~380 KB used

    // 0) trig tables (11,440 precise sin/cos, one tiny launch)
    wealth_build_tables<<<(NUM_NODES + 255) / 256, 256, 0, stream>>>(time_tensor, ws);

    // 1) streamed partial sums on the matrix core: 90 x 50 = 4500 waves
    dim3 grid1(T_TILES, N_CHUNKS);
    wealth_partial_wmma<<<grid1, 32, 0, stream>>>(noise, ws, ws + WS_PARTIAL);

    // 2) finalize 1440 outputs
    wealth_finalize<<<(T_STEPS + 255) / 256, 256, 0, stream>>>(
        time_tensor, ws + WS_PARTIAL, out);
}